// MySDELayer_71244917506160
// MI455X (gfx1250) — compile-verified
//
#include <hip/hip_runtime.h>
#include <hip/hip_bf16.h>

// ---------------- problem constants (match reference) ----------------
#define NUM_TRAJ 1024
#define BATCH    8
#define NTRAJ_ALL (BATCH * NUM_TRAJ)   // 8192
#define NSTEP    255
#define TSTEPS   256                   // includes t=0 initial state
#define DT       0.00390625            // 2^-8
#define OM       5.0265                // P0
#define HG       0.18                  // 0.5 * Ga, Ga = 0.36
// sqrt(Ga/2) * sqrt(DT) = sqrt(0.18) * 0.0625 folded into the noise
#define SGDT     0.026516504294495532

// ---------------- kernel configuration ----------------
#define TPB      64                    // 2 waves/block -> 128 blocks, spread across WGPs
#define WAVES_PB (TPB / 32)
#define CHUNK    5                     // steps per WMMA-reduction chunk (15 columns used)
#define NCHUNK   (NSTEP / CHUNK)       // 51
#define SPITCH   33                    // LDS staging pitch (bank-conflict free)

typedef __attribute__((ext_vector_type(2))) float v2f;
typedef __attribute__((ext_vector_type(8))) float v8f;

// Compiler-only memory fence. The WMMA staging buffer is strictly per-wave and
// CDNA5 LDS instructions from one wave execute in order (DScnt: "in order"),
// so no s_barrier / s_wait_dscnt-0 is needed for the cross-lane store->load
// handoff -- we only need to stop clang from reordering the LDS accesses.
__device__ __forceinline__ void wave_lds_fence() {
    asm volatile("" ::: "memory");
}

// Zero the f32 accumulator in workspace (ws is poisoned by the harness).
__global__ void sde_zero_ws(float* __restrict__ ws, int n) {
    int i = blockIdx.x * blockDim.x + threadIdx.x;
    if (i < n) ws[i] = 0.0f;
}

// One trajectory per lane; f64 state in registers; per-wave mean-reduction
// via V_WMMA_F32_16X16X4_F32 with a ones A-operand (D[m][n] = sum_k B[k][n] + C).
__global__ __launch_bounds__(TPB) void sde_sim(
        const float* __restrict__ inputs,   // (8,1)
        const float* __restrict__ wvec,     // (8192, 255, 2)
        float* __restrict__ S)              // accum: [BATCH][TSTEPS][3] f32
{
    // stage[wave][column 0..15][trajectory-in-wave]; column 15 stays zero
    __shared__ float stage[WAVES_PB][16][SPITCH];

    const int tid  = threadIdx.x;
    const int lane = tid & 31;
    const int wave = tid >> 5;
    const int n    = blockIdx.x * TPB + tid;     // trajectory id 0..8191
    const int b    = n >> 10;                    // output batch = n / 1024

    // parameter: De = inputs[n % 8]  (reference tiles inputs over trajectories)
    const double De = (double)inputs[n & 7];

    // zero the dummy column (never written again; own-wave region only)
    stage[wave][15][lane] = 0.0f;
    wave_lds_fence();

    // state: x0 = (1, 0, 0)
    double x = 1.0, y = 0.0, z = 0.0;

    const float2* wp = (const float2*)wvec + (size_t)n * NSTEP;

    // B-operand gather indices for the WMMA reduction
    const int ncol = lane & 15;                  // column this lane feeds / reads back
    const int half = lane >> 4;                  // which of the 4 K-slots this lane supplies
    const int ft   = ncol / 3;                   // flush: local step  (valid for ncol < 15)
    const int fc   = ncol - 3 * ft;              // flush: component
    float* __restrict__ Sb = S + (size_t)b * TSTEPS * 3;

    int s = 0;                                   // noise step index 0..254
    for (int q = 0; q < NCHUNK; ++q) {
        // pull next chunk's noise (40 contiguous bytes/lane) toward the caches
        if (q + 1 < NCHUNK) __builtin_prefetch(wp + s + CHUNK, 0, 1);

        #pragma unroll
        for (int tl = 0; tl < CHUNK; ++tl, ++s) {
            const float2 w = wp[s];
            const double gi = (double)w.x * SGDT;   // sqrt(Ga/2)*dW_i
            const double gq = (double)w.y * SGDT;   // sqrt(Ga/2)*dW_q

            const double ax = -De * y - HG * x;
            const double ay =  De * x - OM * z - HG * y;
            const double az =  OM * y;

            const double xn = x + DT * ax - x * z * gi + y * gq;
            const double yn = y + DT * ay - y * z * gi - x * gq;
            const double zn = z + DT * az + (1.0 - z * z) * gi;
            x = xn; y = yn; z = zn;

            stage[wave][3 * tl + 0][lane] = (float)x;
            stage[wave][3 * tl + 1][lane] = (float)y;
            stage[wave][3 * tl + 2][lane] = (float)z;
        }
        wave_lds_fence();   // writes above must not sink below the gathers

        // C += Ones(16x4) x B_r(4x16), r = 0..7 covers all 32 trajectories.
        // Column n accumulates sum over trajectories of stage[n][*].
        v8f acc = {};
        v2f ones; ones.x = 1.0f; ones.y = 1.0f;
        #pragma unroll
        for (int r = 0; r < 8; ++r) {
            v2f bmat;
            bmat.x = stage[wave][ncol][4 * r + half];       // K-slot pair 0/1
            bmat.y = stage[wave][ncol][4 * r + 2 + half];   // K-slot pair 2/3
            acc = __builtin_amdgcn_wmma_f32_16x16x4_f32(
                      /*neg_a=*/false, ones, /*neg_b=*/false, bmat,
                      /*c_mod=*/(short)0, acc,
                      /*reuse_a=*/false, /*reuse_b=*/false);
        }
        wave_lds_fence();   // gathers above must complete before next chunk's writes

        // D VGPR0: lanes 0-15 = row M=0, lanes 16-31 = row M=8; all rows equal
        // the column sums, so lane l holds colsum[l % 16].
        const float colsum = acc[0];
        if (lane < 15) {
            const int t = q * CHUNK + 1 + ft;               // absolute timestep 1..255
            atomicAdd(&Sb[t * 3 + fc], colsum);
        }
    }
}

// Convert accumulated sums to the (8, 256, 6) probability output.
__global__ void sde_finalize(const float* __restrict__ S, float* __restrict__ out) {
    const int u = blockIdx.x * blockDim.x + threadIdx.x;   // u = b*256 + t
    if (u >= BATCH * TSTEPS) return;
    const int t = u & (TSTEPS - 1);

    float mx, my, mz;
    if (t == 0) {                       // exact initial state x0 = (1,0,0)
        mx = 1.0f; my = 0.0f; mz = 0.0f;
    } else {
        const float inv = 1.0f / (float)NUM_TRAJ;
        const float* p = S + (size_t)u * 3;
        mx = p[0] * inv; my = p[1] * inv; mz = p[2] * inv;
    }
    const float px = 0.5f * (mx + 1.0f);
    const float py = 0.5f * (my + 1.0f);
    const float pz = 0.5f * (mz + 1.0f);
    float* o = out + (size_t)u * 6;
    o[0] = px; o[1] = 1.0f - px;
    o[2] = py; o[3] = 1.0f - py;
    o[4] = pz; o[5] = 1.0f - pz;
}

extern "C" void kernel_launch(void* const* d_in, const int* in_sizes, int n_in,
                              void* d_out, int out_size, void* d_ws, size_t ws_size,
                              hipStream_t stream) {
    const float* inputs = (const float*)d_in[0];   // (8,1) float32
    const float* wvec   = (const float*)d_in[1];   // (8192,255,2) float32
    float* S   = (float*)d_ws;                     // BATCH*TSTEPS*3 floats accumulator
    float* out = (float*)d_out;                    // (8,256,6) float32

    const int nS = BATCH * TSTEPS * 3;             // 6144
    sde_zero_ws<<<(nS + 255) / 256, 256, 0, stream>>>(S, nS);
    sde_sim<<<NTRAJ_ALL / TPB, TPB, 0, stream>>>(inputs, wvec, S);
    sde_finalize<<<(BATCH * TSTEPS + 255) / 256, 256, 0, stream>>>(S, out);
}